// TimeMix_48026324304446
// MI455X (gfx1250) — compile-verified
//
#include <hip/hip_runtime.h>
#include <math.h>

// ---------------------------------------------------------------------------
// RWKV-6 TimeMix for MI455X (gfx1250).
//  - fp32 WMMA (v_wmma_f32_16x16x4_f32) for all dense projections
//  - global_load_async_to_lds_b128 double-buffered tile staging (ASYNCcnt)
//  - register-resident 64x64 state scan for wkv6
// Shapes fixed by setup_inputs(): B=8, ph=pw=32 -> T=1024, C=768, H=12.
// ---------------------------------------------------------------------------

typedef __attribute__((ext_vector_type(2))) float v2f;
typedef __attribute__((ext_vector_type(8))) float v8f;

#define HEAD 64

enum { ACT_NONE = 0, ACT_RELU = 1, ACT_TANH = 2, ACT_EXPEXP = 3 };

// ---- CDNA5 async-to-LDS helpers -------------------------------------------
__device__ __forceinline__ unsigned lds_addr(const void* p)
{
    // generic -> LDS(3) addrspace cast; AS3 pointers are 32-bit LDS offsets
    return (unsigned)(unsigned long long)(__attribute__((address_space(3))) const void*)p;
}

__device__ __forceinline__ void async_ld_b128(unsigned ldsoff, const float* g)
{
    asm volatile("global_load_async_to_lds_b128 %0, %1, off"
                 :: "v"(ldsoff), "v"(g)
                 : "memory");
}

__device__ __forceinline__ void wait_asynccnt0()
{
    asm volatile("s_wait_asynccnt 0x0" ::: "memory");
}

// ---------------------------------------------------------------------------
// K1: quarter-wise spatial shift (per 64-channel head block), xx = shift(x)-x,
//     xxx = x + xx*maa_x
// ---------------------------------------------------------------------------
__global__ __launch_bounds__(256) void qshift_kernel(
    const float* __restrict__ x, const float* __restrict__ maa_x,
    float* __restrict__ xx, float* __restrict__ xxx,
    int BT, int C, int T, int ph, int pw)
{
    int idx = blockIdx.x * 256 + threadIdx.x;
    if (idx >= BT * C) return;
    int c  = idx % C;
    int bt = idx / C;
    int t  = bt % T;
    int b  = bt / T;
    int hrow = t / pw, wcol = t % pw;
    int quarter = (c % HEAD) >> 4;      // head_dim=64, quarters of 16
    int sh = hrow, sw = wcol;
    bool ok;
    if (quarter == 0)      { sw = wcol - 1; ok = (wcol >= 1);      } // shift right
    else if (quarter == 1) { sw = wcol + 1; ok = (wcol < pw - 1);  } // shift left
    else if (quarter == 2) { sh = hrow - 1; ok = (hrow >= 1);      } // shift down
    else                   { sh = hrow + 1; ok = (hrow < ph - 1);  } // shift up
    float src = 0.f;
    if (ok) src = x[((size_t)b * T + (size_t)sh * pw + sw) * C + c];
    float xv = x[idx];
    float d  = src - xv;
    xx[idx]  = d;
    xxx[idx] = fmaf(d, maa_x[c], xv);
}

// ---------------------------------------------------------------------------
// Generic fp32 WMMA GEMM: out[M,N] = act( A[M,K] @ op(B) + bias[n] )
//   BTMODE==0 : B stored [K,N]  (B[k*N+n])      -> LDS as Bs[k][68-stride]
//   BTMODE==1 : B stored [N,K]  (B[n*K+k], A@B^T) -> LDS as Bs[n][20-stride]
// Block: 256 threads (8 waves), 128x64 tile; each wave -> 32x32 via 4x WMMA
// per k-step (2 A-frags x 2 B-frags). Async double-buffered LDS staging.
// Requires: M % 128 == 0, K % 16 == 0. N may be ragged (sync fallback path).
// ---------------------------------------------------------------------------
#define A_STRIDE 20          // floats per A row (16 + pad), 80B (16B aligned)
#define B0_STRIDE 68         // floats per Bs[k] row (64 + pad), 272B aligned
#define B1_STRIDE 20         // floats per Bs[n] row (16 + pad), 80B aligned
#define A_BUF (128 * A_STRIDE)
#define B_BUF 1280           // max(16*68, 64*20) = 1280 floats

__device__ __forceinline__ void gemm_store(
    float* __restrict__ C, const float* __restrict__ bias,
    int m, int n, int N, float v, int act)
{
    if (n >= N) return;
    if (act == ACT_RELU)        v = fmaxf(v, 0.f);
    else if (act == ACT_TANH)   v = tanhf(v);
    else if (act == ACT_EXPEXP) v = expf(-expf(bias[n] + v));
    C[(size_t)m * N + n] = v;
}

template <int BTMODE>
__global__ __launch_bounds__(256) void gemm_wmma_f32(
    const float* __restrict__ A, const float* __restrict__ B,
    const float* __restrict__ bias, float* __restrict__ Cmat,
    int M, int N, int K, int act)
{
    __shared__ float As[2 * A_BUF];
    __shared__ float Bs[2 * B_BUF];

    const int tid  = threadIdx.x;
    const int lane = tid & 31;
    const int wave = tid >> 5;     // 0..7
    const int wm   = wave >> 1;    // 0..3  -> rows [wm*32, wm*32+32)
    const int wn   = wave & 1;     // 0..1  -> cols [wn*32, wn*32+32)
    const int mBlk = blockIdx.y * 128;
    const int nBlk = blockIdx.x * 64;

    const int row = lane & 15;
    const int kq  = (lane >> 4) << 1;   // lanes 0-15 -> K{0,1}; 16-31 -> K{2,3}
    const bool fullN = (nBlk + 64 <= N);

    const unsigned asBase = lds_addr(&As[0]);
    const unsigned bsBase = lds_addr(&Bs[0]);

    // per-thread staging coordinates (block-shape invariant)
    const int ar = tid >> 1;                 // 0..127  A row
    const int ac = (tid & 1) << 3;           // 0 or 8  A col (two b128s)
    const int b0k = tid >> 4;                // 0..15   (BTMODE==0)
    const int b0n = (tid & 15) << 2;         // 0..60
    const int b1n = tid >> 2;                // 0..63   (BTMODE==1)
    const int b1k = (tid & 3) << 2;          // 0,4,8,12

    v8f acc[2][2];
    acc[0][0] = (v8f){}; acc[0][1] = (v8f){};
    acc[1][0] = (v8f){}; acc[1][1] = (v8f){};

    // ---- stage one 16-wide K chunk into buffer `buf` ----
    auto stage = [&](int k0, int buf) {
        // A tile 128x16 : 2x b128 per thread, async DMA
        {
            const float* ga = A + (size_t)(mBlk + ar) * K + k0 + ac;
            unsigned la = asBase + (unsigned)(buf * A_BUF + ar * A_STRIDE + ac) * 4u;
            async_ld_b128(la, ga);
            async_ld_b128(la + 16u, ga + 4);
        }
        if (BTMODE == 1) {
            // B[n*K+k] rows -> Bs[n][k] (no transpose needed), async DMA
            const float* gb = B + (size_t)(nBlk + b1n) * K + k0 + b1k;
            unsigned lb = bsBase + (unsigned)(buf * B_BUF + b1n * B1_STRIDE + b1k) * 4u;
            async_ld_b128(lb, gb);
        } else if (fullN) {
            // B[k*N+n] rows -> Bs[k][n], async DMA
            const float* gb = B + (size_t)(k0 + b0k) * N + nBlk + b0n;
            unsigned lb = bsBase + (unsigned)(buf * B_BUF + b0k * B0_STRIDE + b0n) * 4u;
            async_ld_b128(lb, gb);
        } else {
            // ragged N: guarded synchronous staging with zero fill
            int gn = nBlk + b0n;
            const float* bp = B + (size_t)(k0 + b0k) * N;
            float4 bv = make_float4(0.f, 0.f, 0.f, 0.f);
            if (gn + 3 < N) {
                bv = *(const float4*)(bp + gn);
            } else {
                if (gn + 0 < N) bv.x = bp[gn + 0];
                if (gn + 1 < N) bv.y = bp[gn + 1];
                if (gn + 2 < N) bv.z = bp[gn + 2];
                if (gn + 3 < N) bv.w = bp[gn + 3];
            }
            float* d = &Bs[buf * B_BUF + b0k * B0_STRIDE + b0n];
            d[0] = bv.x; d[1] = bv.y; d[2] = bv.z; d[3] = bv.w;
        }
    };

    // ---- prologue: fill buffer 0 ----
    stage(0, 0);
    wait_asynccnt0();
    __syncthreads();

    int cur = 0;
    for (int k0 = 0; k0 < K; k0 += 16) {
        if (k0 + 16 < K) stage(k0 + 16, cur ^ 1);   // prefetch next chunk (async)

        const float* __restrict__ Ab = &As[cur * A_BUF];
        const float* __restrict__ Bb = &Bs[cur * B_BUF];

        #pragma unroll
        for (int kk = 0; kk < 16; kk += 4) {
            v2f a0, a1, b0, b1;
            a0.x = Ab[(wm * 32 + row) * A_STRIDE + kk + kq + 0];
            a0.y = Ab[(wm * 32 + row) * A_STRIDE + kk + kq + 1];
            a1.x = Ab[(wm * 32 + 16 + row) * A_STRIDE + kk + kq + 0];
            a1.y = Ab[(wm * 32 + 16 + row) * A_STRIDE + kk + kq + 1];
            if (BTMODE == 1) {
                b0.x = Bb[(wn * 32 + row) * B1_STRIDE + kk + kq + 0];
                b0.y = Bb[(wn * 32 + row) * B1_STRIDE + kk + kq + 1];
                b1.x = Bb[(wn * 32 + 16 + row) * B1_STRIDE + kk + kq + 0];
                b1.y = Bb[(wn * 32 + 16 + row) * B1_STRIDE + kk + kq + 1];
            } else {
                b0.x = Bb[(kk + kq + 0) * B0_STRIDE + wn * 32 + row];
                b0.y = Bb[(kk + kq + 1) * B0_STRIDE + wn * 32 + row];
                b1.x = Bb[(kk + kq + 0) * B0_STRIDE + wn * 32 + 16 + row];
                b1.y = Bb[(kk + kq + 1) * B0_STRIDE + wn * 32 + 16 + row];
            }
            acc[0][0] = __builtin_amdgcn_wmma_f32_16x16x4_f32(
                false, a0, false, b0, (short)0, acc[0][0], false, false);
            acc[0][1] = __builtin_amdgcn_wmma_f32_16x16x4_f32(
                false, a0, false, b1, (short)0, acc[0][1], false, false);
            acc[1][0] = __builtin_amdgcn_wmma_f32_16x16x4_f32(
                false, a1, false, b0, (short)0, acc[1][0], false, false);
            acc[1][1] = __builtin_amdgcn_wmma_f32_16x16x4_f32(
                false, a1, false, b1, (short)0, acc[1][1], false, false);
        }

        wait_asynccnt0();     // our prefetch done (no-op on last iter)
        __syncthreads();      // everyone done reading `cur` + next buffer ready
        cur ^= 1;
    }

    // ---- epilogue: C layout: VGPR v -> M = v (lanes 0-15) / v+8 (16-31) ----
    #pragma unroll
    for (int mi = 0; mi < 2; ++mi) {
        const int mBase = mBlk + wm * 32 + mi * 16 + ((lane >> 4) << 3);
        #pragma unroll
        for (int ni = 0; ni < 2; ++ni) {
            const int n = nBlk + wn * 32 + ni * 16 + (lane & 15);
            #pragma unroll
            for (int v = 0; v < 8; ++v)
                gemm_store(Cmat, bias, mBase + v, n, N, acc[mi][ni][v], act);
        }
    }
}

// ---------------------------------------------------------------------------
// K3: m_f = t5 @ maa_w2[f]  (K=32 per f), fused with the 5 mix writes.
// One block per token row m; t5 row cached in LDS.
// ---------------------------------------------------------------------------
__global__ __launch_bounds__(256) void mix_kernel(
    const float* __restrict__ x, const float* __restrict__ xx,
    const float* __restrict__ t5, const float* __restrict__ w2,
    const float* __restrict__ maa_w, const float* __restrict__ maa_k,
    const float* __restrict__ maa_v, const float* __restrict__ maa_r,
    const float* __restrict__ maa_g,
    float* __restrict__ xw, float* __restrict__ xk, float* __restrict__ xv,
    float* __restrict__ xr, float* __restrict__ xg, int C)
{
    __shared__ float trow[160];
    const int m   = blockIdx.x;
    const int tid = threadIdx.x;
    if (tid < 160) trow[tid] = t5[(size_t)m * 160 + tid];
    __syncthreads();

    for (int c = tid; c < C; c += 256) {
        float s0 = 0.f, s1 = 0.f, s2 = 0.f, s3 = 0.f, s4 = 0.f;
        #pragma unroll 4
        for (int r = 0; r < 32; ++r) {
            s0 = fmaf(trow[0 * 32 + r], w2[(size_t)(0 * 32 + r) * C + c], s0);
            s1 = fmaf(trow[1 * 32 + r], w2[(size_t)(1 * 32 + r) * C + c], s1);
            s2 = fmaf(trow[2 * 32 + r], w2[(size_t)(2 * 32 + r) * C + c], s2);
            s3 = fmaf(trow[3 * 32 + r], w2[(size_t)(3 * 32 + r) * C + c], s3);
            s4 = fmaf(trow[4 * 32 + r], w2[(size_t)(4 * 32 + r) * C + c], s4);
        }
        size_t base = (size_t)m * C + c;
        float xb = x[base];
        float d  = xx[base];
        xw[base] = fmaf(d, maa_w[c] + s0, xb);
        xk[base] = fmaf(d, maa_k[c] + s1, xb);
        xv[base] = fmaf(d, maa_v[c] + s2, xb);
        xr[base] = fmaf(d, maa_r[c] + s3, xb);
        xg[base] = fmaf(d, maa_g[c] + s4, xb);
    }
}

// ---------------------------------------------------------------------------
// K7: wkv6 linear recurrence. One block per (b,h). 256 threads:
//   thread -> (i = tid&63 value lane, jg = tid>>6 key-quarter), S[16] in regs.
//   wdec already holds exp(-exp(time_decay + ww)).
// ---------------------------------------------------------------------------
__global__ __launch_bounds__(256) void wkv6_kernel(
    const float* __restrict__ r, const float* __restrict__ k,
    const float* __restrict__ v, const float* __restrict__ wdec,
    const float* __restrict__ u, float* __restrict__ y,
    int B, int T, int C, int H)
{
    const int bh  = blockIdx.x;
    const int b   = bh / H;
    const int h   = bh % H;
    const int tid = threadIdx.x;
    const int i   = tid & 63;
    const int jg  = tid >> 6;          // 0..3

    __shared__ float rs[64], ks[64], ws[64], vs[64], us[64];
    __shared__ float red[4][64];

    if (tid < 64) us[tid] = u[h * HEAD + tid];

    float S[16];
    #pragma unroll
    for (int z = 0; z < 16; ++z) S[z] = 0.f;
    __syncthreads();

    const size_t baseBH = (size_t)b * T * C + (size_t)h * HEAD;
    for (int t = 0; t < T; ++t) {
        size_t base = baseBH + (size_t)t * C;
        if (tid < 64)       rs[tid]       = r[base + tid];
        else if (tid < 128) ks[tid - 64]  = k[base + tid - 64];
        else if (tid < 192) ws[tid - 128] = wdec[base + tid - 128];
        else                vs[tid - 192] = v[base + tid - 192];
        __syncthreads();

        float vi = vs[i];
        float p  = 0.f;
        #pragma unroll
        for (int jj = 0; jj < 16; ++jj) {
            int   j  = jg * 16 + jj;
            float kv = ks[j] * vi;
            p    = fmaf(rs[j], fmaf(us[j], kv, S[jj]), p);
            S[jj] = fmaf(ws[j], S[jj], kv);
        }
        red[jg][i] = p;
        __syncthreads();
        if (jg == 0)
            y[base + i] = red[0][i] + red[1][i] + red[2][i] + red[3][i];
    }
}

// ---------------------------------------------------------------------------
// K8: GroupNorm over 64-channel head groups, times g. One block per (m,h).
// ---------------------------------------------------------------------------
__global__ __launch_bounds__(64) void gn_mul_kernel(
    const float* __restrict__ y, const float* __restrict__ g,
    const float* __restrict__ ln_w, const float* __restrict__ ln_b,
    float* __restrict__ z, int C, int H)
{
    const int blk = blockIdx.x;
    const int m   = blk / H;
    const int h   = blk % H;
    const int tid = threadIdx.x;

    __shared__ float ssum[64], ssq[64];
    const size_t base = (size_t)m * C + (size_t)h * HEAD;
    float val = y[base + tid];
    ssum[tid] = val;
    ssq[tid]  = val * val;
    __syncthreads();
    #pragma unroll
    for (int off = 32; off > 0; off >>= 1) {
        if (tid < off) {
            ssum[tid] += ssum[tid + off];
            ssq[tid]  += ssq[tid + off];
        }
        __syncthreads();
    }
    float mu  = ssum[0] * (1.f / 64.f);
    float var = ssq[0] * (1.f / 64.f) - mu * mu;
    float inv = rsqrtf(var + 1e-5f);
    int   c   = h * HEAD + tid;
    float nv  = (val - mu) * inv * ln_w[c] + ln_b[c];
    z[base + tid] = nv * g[base + tid];
}

// ---------------------------------------------------------------------------
// Host-side orchestration
// ---------------------------------------------------------------------------
extern "C" void kernel_launch(void* const* d_in, const int* in_sizes, int n_in,
                              void* d_out, int out_size, void* d_ws, size_t ws_size,
                              hipStream_t stream)
{
    const float* x     = (const float*)d_in[0];
    const float* W_r   = (const float*)d_in[1];
    const float* W_k   = (const float*)d_in[2];
    const float* W_v   = (const float*)d_in[3];
    const float* W_g   = (const float*)d_in[4];
    const float* W_o   = (const float*)d_in[5];
    const float* maa_x = (const float*)d_in[6];
    const float* maa_w = (const float*)d_in[7];
    const float* maa_k = (const float*)d_in[8];
    const float* maa_v = (const float*)d_in[9];
    const float* maa_r = (const float*)d_in[10];
    const float* maa_g = (const float*)d_in[11];
    const float* maa_w1= (const float*)d_in[12];
    const float* maa_w2= (const float*)d_in[13];
    const float* tdec  = (const float*)d_in[14];
    const float* dw1   = (const float*)d_in[15];
    const float* dw2   = (const float*)d_in[16];
    const float* faaaa = (const float*)d_in[17];
    const float* ln_w  = (const float*)d_in[18];
    const float* ln_b  = (const float*)d_in[19];

    const int C  = in_sizes[6];          // 768
    const int BT = in_sizes[0] / C;      // 8192
    const int PH = 32, PW = 32;          // per setup_inputs()
    const int T  = PH * PW;              // 1024
    const int B  = BT / T;               // 8
    const int H  = C / HEAD;             // 12

    float* ws = (float*)d_ws;
    const size_t big = (size_t)BT * C;
    float* XX = ws;              // xx
    float* XB = XX + big;        // xxx -> xw -> w(decayed)
    float* XK = XB + big;        // xk  -> y
    float* XV = XK + big;        // xv  -> z
    float* XR = XV + big;        // xr
    float* XG = XR + big;        // xg
    float* Rb = XG + big;
    float* Kb = Rb + big;
    float* Vb = Kb + big;
    float* Gb = Vb + big;
    float* T5 = Gb + big;                  // BT x 160
    float* H1 = T5 + (size_t)BT * 160;     // BT x 64
    (void)ws_size; (void)n_in; (void)out_size;

    const dim3 blk256(256);
    const int  my = BT / 128;             // 128-row block tiles

    // K1: shift + xx/xxx
    qshift_kernel<<<(BT * C + 255) / 256, blk256, 0, stream>>>(
        x, maa_x, XX, XB, BT, C, T, PH, PW);

    // K2: t5 = tanh(xxx @ maa_w1)          [BT,160]
    gemm_wmma_f32<0><<<dim3((160 + 63) / 64, my), blk256, 0, stream>>>(
        XB, maa_w1, nullptr, T5, BT, 160, C, ACT_TANH);

    // K3: mix -> xw,xk,xv,xr,xg (xw overwrites xxx in XB)
    mix_kernel<<<BT, blk256, 0, stream>>>(
        x, XX, T5, maa_w2, maa_w, maa_k, maa_v, maa_r, maa_g,
        XB, XK, XV, XR, XG, C);

    // K4: r,k,v,g projections (weights stored [out,in] -> A @ W^T)
    gemm_wmma_f32<1><<<dim3(C / 64, my), blk256, 0, stream>>>(
        XR, W_r, nullptr, Rb, BT, C, C, ACT_NONE);
    gemm_wmma_f32<1><<<dim3(C / 64, my), blk256, 0, stream>>>(
        XK, W_k, nullptr, Kb, BT, C, C, ACT_NONE);
    gemm_wmma_f32<1><<<dim3(C / 64, my), blk256, 0, stream>>>(
        XV, W_v, nullptr, Vb, BT, C, C, ACT_NONE);
    gemm_wmma_f32<1><<<dim3(C / 64, my), blk256, 0, stream>>>(
        XG, W_g, nullptr, Gb, BT, C, C, ACT_RELU);

    // K5: h1 = tanh(xw @ dec_w1)           [BT,64]
    gemm_wmma_f32<0><<<dim3(1, my), blk256, 0, stream>>>(
        XB, dw1, nullptr, H1, BT, 64, C, ACT_TANH);

    // K6: w = exp(-exp(time_decay + h1 @ dec_w2))   (overwrites XB)
    gemm_wmma_f32<0><<<dim3(C / 64, my), blk256, 0, stream>>>(
        H1, dw2, tdec, XB, BT, C, 64, ACT_EXPEXP);

    // K7: wkv6 scan -> y (in XK)
    wkv6_kernel<<<B * H, blk256, 0, stream>>>(
        Rb, Kb, Vb, XB, faaaa, XK, B, T, C, H);

    // K8: groupnorm * g -> z (in XV)
    gn_mul_kernel<<<BT * H, dim3(64), 0, stream>>>(
        XK, Gb, ln_w, ln_b, XV, C, H);

    // K9: out = z @ W_o.T
    gemm_wmma_f32<1><<<dim3(C / 64, my), blk256, 0, stream>>>(
        XV, W_o, nullptr, (float*)d_out, BT, C, C, ACT_NONE);
}